// GraphAttentionEncoder_46986942218774
// MI455X (gfx1250) — compile-verified
//
#include <hip/hip_runtime.h>
#include <hip/hip_bf16.h>

typedef __attribute__((ext_vector_type(2))) float v2f;
typedef __attribute__((ext_vector_type(8))) float v8f;

#define NEG_SLOPE 0.2f
#define BN_EPS 1e-5f

// ------------------------------------------------------------------
// helpers
// ------------------------------------------------------------------
__device__ __forceinline__ float leaky(float v) { return v > 0.f ? v : NEG_SLOPE * v; }

// monotone order-preserving float<->uint mapping for atomicMax on floats
__device__ __forceinline__ unsigned flipf(float f) {
  unsigned u = __float_as_uint(f);
  return (u & 0x80000000u) ? ~u : (u | 0x80000000u);
}
__device__ __forceinline__ float unflipf(unsigned u) {
  return __uint_as_float((u & 0x80000000u) ? (u & 0x7FFFFFFFu) : ~u);
}

// edge e in [0, E+n): first E are edge_index, last n are self loops
__device__ __forceinline__ void edge_sd(const int* __restrict__ ei, int E, int e,
                                        int& s, int& d) {
  if (e < E) { s = ei[e]; d = ei[E + e]; }
  else       { s = e - E; d = e - E; }
}

// ------------------------------------------------------------------
// fp32 WMMA GEMM:  C[M,Nc] = A[M,K] * B[K,Nc]   (one 16x16 tile / wave)
// V_WMMA_F32_16X16X4_F32, K stepped by 4.
// ------------------------------------------------------------------
__global__ void wmma_gemm_f32(const float* __restrict__ A,
                              const float* __restrict__ B,
                              float* __restrict__ C,
                              int M, int K, int Nc, int tilesN, int totalTiles) {
  int wave = blockIdx.x * (blockDim.x >> 5) + (threadIdx.x >> 5);
  if (wave >= totalTiles) return;               // wave-uniform: EXEC stays all-ones
  int lane = threadIdx.x & 31;
  int half = lane >> 4;                          // 0: K=0,1   1: K=2,3
  int l    = lane & 15;
  int tM = wave / tilesN;
  int tN = wave % tilesN;

  int rowA = tM * 16 + l;
  if (rowA > M - 1) rowA = M - 1;                // clamp (M % 16 == 0 in practice)
  const float* arow = A + (size_t)rowA * K;
  const float* bcol = B + tN * 16 + l;

  v8f acc = {};
  for (int k = 0; k < K; k += 4) {
    int ka = k + half * 2;
    v2f a, b;
    a.x = arow[ka];
    a.y = arow[ka + 1];
    const float* bp = bcol + (size_t)ka * Nc;
    b.x = bp[0];
    b.y = bp[Nc];
    acc = __builtin_amdgcn_wmma_f32_16x16x4_f32(false, a, false, b,
                                                (short)0, acc, false, false);
  }

  int rbase = tM * 16 + half * 8;
  float* cp = C + (size_t)rbase * Nc + tN * 16 + l;
#pragma unroll
  for (int r = 0; r < 8; ++r)
    if (rbase + r < M) cp[(size_t)r * Nc] = acc[r];
}

// ------------------------------------------------------------------
// workspace init (graph-replay safe: runs every call)
// ------------------------------------------------------------------
__global__ void init_ws(float* agg1, float* agg2, float* s1, float* s2,
                        float* stats1, float* stats2, float* gsum,
                        unsigned* gmaxb, float* cnt, int n, int Bg) {
  int i = blockIdx.x * blockDim.x + threadIdx.x;
  if (i < n * 64) agg1[i] = 0.f;
  if (i < n * 32) agg2[i] = 0.f;
  if (i < n * 4)  s1[i] = 0.f;
  if (i < n)      s2[i] = 0.f;
  if (i < 128)    stats1[i] = 0.f;
  if (i < 64)     stats2[i] = 0.f;
  if (i < Bg * 32) { gsum[i] = 0.f; gmaxb[i] = 0x007FFFFFu; }   // flip(-inf)
  if (i < Bg)     cnt[i] = 0.f;
}

// ------------------------------------------------------------------
// per-node attention logits
// ------------------------------------------------------------------
__global__ void attn_coeff_l1(const float* __restrict__ xl,
                              const float* __restrict__ a_src,
                              const float* __restrict__ a_dst,
                              float* __restrict__ es, float* __restrict__ ed, int n) {
  int t = blockIdx.x * blockDim.x + threadIdx.x;
  if (t >= n * 4) return;
  int node = t >> 2, h = t & 3;
  const float* row = xl + (size_t)node * 64 + h * 16;
  float s = 0.f, d = 0.f;
#pragma unroll
  for (int c = 0; c < 16; ++c) {
    s += row[c] * a_src[h * 16 + c];
    d += row[c] * a_dst[h * 16 + c];
  }
  es[t] = s; ed[t] = d;
}

__global__ void attn_coeff_l2(const float* __restrict__ h2,
                              const float* __restrict__ a_src,
                              const float* __restrict__ a_dst,
                              float* __restrict__ es, float* __restrict__ ed, int n) {
  int t = blockIdx.x * blockDim.x + threadIdx.x;
  if (t >= n) return;
  const float* row = h2 + (size_t)t * 32;
  float s = 0.f, d = 0.f;
#pragma unroll
  for (int c = 0; c < 32; ++c) {
    s += row[c] * a_src[c];
    d += row[c] * a_dst[c];
  }
  es[t] = s; ed[t] = d;
}

// ------------------------------------------------------------------
// edge softmax denominators (max dropped: softmax is shift-invariant,
// logits here are O(1) so exp() is safe without it)
// ------------------------------------------------------------------
__global__ void edge_sum_l1(const int* __restrict__ ei, int E, int Etot,
                            const float* __restrict__ es, const float* __restrict__ ed,
                            float* __restrict__ s1) {
  int e = blockIdx.x * blockDim.x + threadIdx.x;
  if (e >= Etot) return;
  int s, d; edge_sd(ei, E, e, s, d);
#pragma unroll
  for (int h = 0; h < 4; ++h) {
    float p = expf(leaky(es[s * 4 + h] + ed[d * 4 + h]));
    atomicAdd(&s1[d * 4 + h], p);
  }
}

__global__ void edge_sum_l2(const int* __restrict__ ei, int E, int Etot,
                            const float* __restrict__ es, const float* __restrict__ ed,
                            float* __restrict__ s2) {
  int e = blockIdx.x * blockDim.x + threadIdx.x;
  if (e >= Etot) return;
  int s, d; edge_sd(ei, E, e, s, d);
  float p = expf(leaky(es[s] + ed[d]));
  atomicAdd(&s2[d], p);
}

// ------------------------------------------------------------------
// edge aggregation: one thread per (edge, channel) -> contiguous gathers
// ------------------------------------------------------------------
__global__ void edge_agg_l1(const int* __restrict__ ei, int E, int Etot,
                            const float* __restrict__ es, const float* __restrict__ ed,
                            const float* __restrict__ s1,
                            const float* __restrict__ xl, float* __restrict__ agg1) {
  int t = blockIdx.x * blockDim.x + threadIdx.x;
  if (t >= Etot * 64) return;
  int e = t >> 6, ch = t & 63, h = ch >> 4;
  int s, d; edge_sd(ei, E, e, s, d);
  float p = expf(leaky(es[s * 4 + h] + ed[d * 4 + h]));
  float alpha = p / (s1[d * 4 + h] + 1e-16f);
  atomicAdd(&agg1[(size_t)d * 64 + ch], alpha * xl[(size_t)s * 64 + ch]);
}

__global__ void edge_agg_l2(const int* __restrict__ ei, int E, int Etot,
                            const float* __restrict__ es, const float* __restrict__ ed,
                            const float* __restrict__ s2,
                            const float* __restrict__ h2, float* __restrict__ agg2) {
  int t = blockIdx.x * blockDim.x + threadIdx.x;
  if (t >= Etot * 32) return;
  int e = t >> 5, ch = t & 31;
  int s, d; edge_sd(ei, E, e, s, d);
  float p = expf(leaky(es[s] + ed[d]));
  float alpha = p / (s2[d] + 1e-16f);
  atomicAdd(&agg2[(size_t)d * 32 + ch], alpha * h2[(size_t)s * 32 + ch]);
}

// ------------------------------------------------------------------
// batch-norm statistics (sum + sumsq per feature); bias folded in
// requires: f divides blockDim (256) -> col fixed per thread
// ------------------------------------------------------------------
__global__ void bn_stats(const float* __restrict__ h, const float* __restrict__ bias,
                         float* __restrict__ stats, int n, int f) {
  __shared__ float ss[256], sq[256];
  int idx0 = blockIdx.x * blockDim.x + threadIdx.x;
  int stride = blockDim.x * gridDim.x;           // multiple of f
  int total = n * f;
  int col = threadIdx.x % f;
  float bv = bias[col];
  float a = 0.f, b = 0.f;
  for (int i = idx0; i < total; i += stride) {
    float v = h[i] + bv;
    a += v; b += v * v;
  }
  ss[threadIdx.x] = a; sq[threadIdx.x] = b;
  __syncthreads();
  if (threadIdx.x < f) {
    float sa = 0.f, sb = 0.f;
    for (int j = threadIdx.x; j < 256; j += f) { sa += ss[j]; sb += sq[j]; }
    atomicAdd(&stats[threadIdx.x], sa);
    atomicAdd(&stats[f + threadIdx.x], sb);
  }
}

__global__ void bn_relu_l1(float* __restrict__ h, const float* __restrict__ bias,
                           const float* __restrict__ stats,
                           const float* __restrict__ gamma, const float* __restrict__ beta,
                           int n) {
  int i = blockIdx.x * blockDim.x + threadIdx.x;
  if (i >= n * 64) return;
  int col = i & 63;
  float invn = 1.f / (float)n;
  float mu = stats[col] * invn;
  float var = stats[64 + col] * invn - mu * mu;
  float v = h[i] + bias[col];
  float y = (v - mu) * rsqrtf(var + BN_EPS) * gamma[col] + beta[col];
  h[i] = fmaxf(y, 0.f);
}

// BN layer2 -> node_emb out, plus graph pooling atomics (mean-sum, max, count)
__global__ void bn_pool_l2(const float* __restrict__ agg2, const float* __restrict__ bias,
                           const float* __restrict__ stats,
                           const float* __restrict__ gamma, const float* __restrict__ beta,
                           const int* __restrict__ batch,
                           float* __restrict__ node_emb,
                           float* __restrict__ gsum, unsigned* __restrict__ gmaxb,
                           float* __restrict__ cnt, int n) {
  int i = blockIdx.x * blockDim.x + threadIdx.x;
  if (i >= n * 32) return;
  int col = i & 31, node = i >> 5;
  float invn = 1.f / (float)n;
  float mu = stats[col] * invn;
  float var = stats[32 + col] * invn - mu * mu;
  float v = agg2[i] + bias[col];
  float y = (v - mu) * rsqrtf(var + BN_EPS) * gamma[col] + beta[col];
  node_emb[i] = y;
  int b = batch[node];
  atomicAdd(&gsum[b * 32 + col], y);
  atomicMax(&gmaxb[b * 32 + col], flipf(y));
  if (col == 0) atomicAdd(&cnt[b], 1.f);
}

// ------------------------------------------------------------------
// graph head: relu(concat(mean,max) @ fcW1 + fcb1) @ fcW2 + fcb2
// ------------------------------------------------------------------
__global__ void fc1_k(const float* __restrict__ gsum, const unsigned* __restrict__ gmaxb,
                      const float* __restrict__ cnt,
                      const float* __restrict__ W, const float* __restrict__ bvec,
                      float* __restrict__ out, int Bg) {
  int t = blockIdx.x * blockDim.x + threadIdx.x;
  if (t >= Bg * 32) return;
  int b = t >> 5, c = t & 31;
  float denom = fmaxf(cnt[b], 1.f);
  float acc = bvec[c];
#pragma unroll
  for (int k = 0; k < 32; ++k)
    acc += (gsum[b * 32 + k] / denom) * W[k * 32 + c];
#pragma unroll
  for (int k = 0; k < 32; ++k) {
    float m = unflipf(gmaxb[b * 32 + k]);
    if (!(m > -3.0e38f && m < 3.0e38f)) m = 0.f;   // isfinite guard
    acc += m * W[(32 + k) * 32 + c];
  }
  out[t] = fmaxf(acc, 0.f);
}

__global__ void fc2_k(const float* __restrict__ gin, const float* __restrict__ W,
                      const float* __restrict__ bvec, float* __restrict__ out, int Bg) {
  int t = blockIdx.x * blockDim.x + threadIdx.x;
  if (t >= Bg * 32) return;
  int b = t >> 5, c = t & 31;
  float acc = bvec[c];
#pragma unroll
  for (int k = 0; k < 32; ++k) acc += gin[b * 32 + k] * W[k * 32 + c];
  out[t] = acc;
}

// ------------------------------------------------------------------
extern "C" void kernel_launch(void* const* d_in, const int* in_sizes, int n_in,
                              void* d_out, int out_size, void* d_ws, size_t ws_size,
                              hipStream_t stream) {
  const float* x      = (const float*)d_in[0];
  const int*   ei     = (const int*)d_in[1];
  const int*   batch  = (const int*)d_in[2];
  const float* W1     = (const float*)d_in[3];
  const float* a_src1 = (const float*)d_in[4];
  const float* a_dst1 = (const float*)d_in[5];
  const float* b1     = (const float*)d_in[6];
  const float* g1     = (const float*)d_in[7];
  const float* be1    = (const float*)d_in[8];
  const float* W2     = (const float*)d_in[9];
  const float* a_src2 = (const float*)d_in[10];
  const float* a_dst2 = (const float*)d_in[11];
  const float* b2     = (const float*)d_in[12];
  const float* g2     = (const float*)d_in[13];
  const float* be2    = (const float*)d_in[14];
  const float* fcW1   = (const float*)d_in[15];
  const float* fcb1   = (const float*)d_in[16];
  const float* fcW2   = (const float*)d_in[17];
  const float* fcb2   = (const float*)d_in[18];

  const int n    = in_sizes[0] / 128;            // 50000
  const int E    = in_sizes[1] / 2;              // 800000
  const int Etot = E + n;                        // + self loops
  const int Bg   = (out_size - n * 32) / 32;     // 64 graphs

  // ---- workspace layout (floats) ----
  float* ws     = (float*)d_ws;
  float* xl     = ws;                            // [n,64]
  float* agg1   = xl   + (size_t)n * 64;         // [n,64] (becomes h1 in place)
  float* h2     = agg1 + (size_t)n * 64;         // [n,32]
  float* agg2   = h2   + (size_t)n * 32;         // [n,32]
  float* es1    = agg2 + (size_t)n * 32;         // [n,4]
  float* ed1    = es1  + (size_t)n * 4;
  float* s1     = ed1  + (size_t)n * 4;
  float* es2    = s1   + (size_t)n * 4;          // [n]
  float* ed2    = es2  + n;
  float* s2     = ed2  + n;
  float* stats1 = s2   + n;                      // [128]
  float* stats2 = stats1 + 128;                  // [64]
  float* gsum   = stats2 + 64;                   // [Bg,32]
  unsigned* gmaxb = (unsigned*)(gsum + (size_t)Bg * 32);
  float* cnt    = (float*)(gmaxb + (size_t)Bg * 32);
  float* g1out  = cnt + Bg;                      // [Bg,32]

  float* node_emb  = (float*)d_out;
  float* graph_emb = (float*)d_out + (size_t)n * 32;

  const int TB = 256;
  auto cdiv = [](long long a, long long b) { return (int)((a + b - 1) / b); };

  // 0) init accumulators
  init_ws<<<cdiv((long long)n * 64, TB), TB, 0, stream>>>(
      agg1, agg2, s1, s2, stats1, stats2, gsum, gmaxb, cnt, n, Bg);

  // 1) xl = x @ W1   [n,128]x[128,64]
  {
    int tilesM = (n + 15) / 16, tilesN = 4;
    int total = tilesM * tilesN;
    wmma_gemm_f32<<<cdiv(total, 8), TB, 0, stream>>>(x, W1, xl, n, 128, 64, tilesN, total);
  }

  // 2) attention logits, softmax denominators, aggregation (layer 1)
  attn_coeff_l1<<<cdiv((long long)n * 4, TB), TB, 0, stream>>>(xl, a_src1, a_dst1, es1, ed1, n);
  edge_sum_l1<<<cdiv(Etot, TB), TB, 0, stream>>>(ei, E, Etot, es1, ed1, s1);
  edge_agg_l1<<<cdiv((long long)Etot * 64, TB), TB, 0, stream>>>(ei, E, Etot, es1, ed1, s1, xl, agg1);

  // 3) BN + ReLU (bias folded in); h1 overwrites agg1
  bn_stats<<<256, TB, 0, stream>>>(agg1, b1, stats1, n, 64);
  bn_relu_l1<<<cdiv((long long)n * 64, TB), TB, 0, stream>>>(agg1, b1, stats1, g1, be1, n);

  // 4) h2 = h1 @ W2   [n,64]x[64,32]
  {
    int tilesM = (n + 15) / 16, tilesN = 2;
    int total = tilesM * tilesN;
    wmma_gemm_f32<<<cdiv(total, 8), TB, 0, stream>>>(agg1, W2, h2, n, 64, 32, tilesN, total);
  }

  // 5) layer 2 attention + aggregation (single head, C=32)
  attn_coeff_l2<<<cdiv(n, TB), TB, 0, stream>>>(h2, a_src2, a_dst2, es2, ed2, n);
  edge_sum_l2<<<cdiv(Etot, TB), TB, 0, stream>>>(ei, E, Etot, es2, ed2, s2);
  edge_agg_l2<<<cdiv((long long)Etot * 32, TB), TB, 0, stream>>>(ei, E, Etot, es2, ed2, s2, h2, agg2);

  // 6) BN -> node_emb, fused graph pooling (sum / max / count)
  bn_stats<<<256, TB, 0, stream>>>(agg2, b2, stats2, n, 32);
  bn_pool_l2<<<cdiv((long long)n * 32, TB), TB, 0, stream>>>(
      agg2, b2, stats2, g2, be2, batch, node_emb, gsum, gmaxb, cnt, n);

  // 7) graph head MLP
  fc1_k<<<cdiv((long long)Bg * 32, TB), TB, 0, stream>>>(gsum, gmaxb, cnt, fcW1, fcb1, g1out, Bg);
  fc2_k<<<cdiv((long long)Bg * 32, TB), TB, 0, stream>>>(g1out, fcW2, fcb2, graph_emb, Bg);
}